// Gaussian_Splatter_88313117540420
// MI455X (gfx1250) — compile-verified
//
#include <hip/hip_runtime.h>

// Gaussian splat forward for MI455X (gfx1250, wave32).
// Bandwidth-bound (~200 MB @ 23.3 TB/s ~= 8.6 us floor). Camera transform
// routed through V_WMMA_F32_16X16X4_F32: D = [rot|tran](16x4) x homPos(4x16).
// All divisions / sqrt / rsqrt use raw 1-ulp hardware approximation ops
// (v_rcp_f32 / v_sqrt_f32 / v_rsq_f32) to keep VALU count below the HBM floor.

typedef float v2f __attribute__((ext_vector_type(2)));
typedef float v8f __attribute__((ext_vector_type(8)));

#define GS_EPS 0.0001f

__device__ __forceinline__ float fast_rcp(float x)  { return __builtin_amdgcn_rcpf(x); }
__device__ __forceinline__ float fast_sqrt(float x) { return __builtin_amdgcn_sqrtf(x); }
__device__ __forceinline__ float fast_rsq(float x)  { return __builtin_amdgcn_rsqf(x); }

__device__ __forceinline__ float fast_sigmoid(float x) {
    // 1 / (1 + e^-x); denominator in [1, inf) -> raw v_rcp_f32 is plenty.
    return fast_rcp(1.0f + __expf(-x));
}

// Everything after pos_cam (shared by WMMA kernel and scalar tail kernel).
__device__ __forceinline__ void gs_finish(
    int tid, float xc, float yc, float zc,
    float r00, float r01, float r02,
    float r10, float r11, float r12,
    float r20, float r21, float r22,
    const float* __restrict__ rgb,
    const float* __restrict__ opacity,
    const float* __restrict__ quat,
    const float* __restrict__ scale,
    float* __restrict__ out_pos,
    float* __restrict__ out_cov,
    float* __restrict__ out_rgb,
    float* __restrict__ out_op)
{
    const float len = fast_sqrt(xc * xc + yc * yc + zc * zc);

    // 1/z with one Newton step (feeds iz^2 in J, keep it tight).
    float iz = fast_rcp(zc);
    iz = iz * (2.0f - zc * iz);

    out_pos[3 * tid + 0] = xc * iz;
    out_pos[3 * tid + 1] = yc * iz;
    out_pos[3 * tid + 2] = len;

    // quaternion (w,x,y,z) -> rotation matrix
    const float4 q = reinterpret_cast<const float4*>(quat)[tid];
    const float qinv = fast_rsq(q.x * q.x + q.y * q.y + q.z * q.z + q.w * q.w);
    const float w = q.x * qinv, x = q.y * qinv, y = q.z * qinv, z = q.w * qinv;

    const float R00 = 1.0f - 2.0f * (y * y + z * z);
    const float R01 = 2.0f * (x * y - w * z);
    const float R02 = 2.0f * (x * z + w * y);
    const float R10 = 2.0f * (x * y + w * z);
    const float R11 = 1.0f - 2.0f * (x * x + z * z);
    const float R12 = 2.0f * (y * z - w * x);
    const float R20 = 2.0f * (x * z - w * y);
    const float R21 = 2.0f * (y * z + w * x);
    const float R22 = 1.0f - 2.0f * (x * x + y * y);

    const float s0 = fabsf(scale[3 * tid + 0]) + GS_EPS;
    const float s1 = fabsf(scale[3 * tid + 1]) + GS_EPS;
    const float s2 = fabsf(scale[3 * tid + 2]) + GS_EPS;

    // RS = R * diag(s)
    const float A00 = R00 * s0, A01 = R01 * s1, A02 = R02 * s2;
    const float A10 = R10 * s0, A11 = R11 * s1, A12 = R12 * s2;
    const float A20 = R20 * s0, A21 = R21 * s1, A22 = R22 * s2;

    // cov3d = RS * RS^T (symmetric)
    const float c00 = A00 * A00 + A01 * A01 + A02 * A02;
    const float c01 = A00 * A10 + A01 * A11 + A02 * A12;
    const float c02 = A00 * A20 + A01 * A21 + A02 * A22;
    const float c11 = A10 * A10 + A11 * A11 + A12 * A12;
    const float c12 = A10 * A20 + A11 * A21 + A12 * A22;
    const float c22 = A20 * A20 + A21 * A21 + A22 * A22;

    // J rows 0,1 only (row 2 never reaches cov2d[:2,:2]);
    // JW row i = J[i][i]*rot_row_i + J[i][2]*rot_row_2
    const float iz2 = iz * iz;
    const float J02 = -xc * iz2;
    const float J12 = -yc * iz2;

    const float W00 = iz * r00 + J02 * r20;
    const float W01 = iz * r01 + J02 * r21;
    const float W02 = iz * r02 + J02 * r22;
    const float W10 = iz * r10 + J12 * r20;
    const float W11 = iz * r11 + J12 * r21;
    const float W12 = iz * r12 + J12 * r22;

    // T = W(2x3) * cov3d(3x3)
    const float T00 = W00 * c00 + W01 * c01 + W02 * c02;
    const float T01 = W00 * c01 + W01 * c11 + W02 * c12;
    const float T02 = W00 * c02 + W01 * c12 + W02 * c22;
    const float T10 = W10 * c00 + W11 * c01 + W12 * c02;
    const float T11 = W10 * c01 + W11 * c11 + W12 * c12;
    const float T12 = W10 * c02 + W11 * c12 + W12 * c22;

    // cov2d = T * W^T; (1,0) == (0,1) exactly in math (cov3d symmetric).
    const float v00 = T00 * W00 + T01 * W01 + T02 * W02;
    const float v01 = T00 * W10 + T01 * W11 + T02 * W12;
    const float v11 = T10 * W10 + T11 * W11 + T12 * W12;
    out_cov[4 * tid + 0] = v00;
    out_cov[4 * tid + 1] = v01;
    out_cov[4 * tid + 2] = v01;
    out_cov[4 * tid + 3] = v11;

    out_rgb[3 * tid + 0] = fast_sigmoid(rgb[3 * tid + 0]);
    out_rgb[3 * tid + 1] = fast_sigmoid(rgb[3 * tid + 1]);
    out_rgb[3 * tid + 2] = fast_sigmoid(rgb[3 * tid + 2]);
    out_op[tid]          = fast_sigmoid(opacity[tid]);
}

// Main kernel: one point per lane; camera transform via WMMA f32 16x16x4.
// Launched only on a multiple of 128 points -> every wave fully active
// (EXEC all-ones, required for WMMA).
__global__ __launch_bounds__(128) void gs_fwd_wmma(
    const float* __restrict__ position,
    const float* __restrict__ rgb,
    const float* __restrict__ opacity,
    const float* __restrict__ quat,
    const float* __restrict__ scale,
    const float* __restrict__ rot,
    const float* __restrict__ tran,
    float* __restrict__ out_pos,
    float* __restrict__ out_cov,
    float* __restrict__ out_rgb,
    float* __restrict__ out_op)
{
    const int tid  = blockIdx.x * 128 + threadIdx.x;
    const int lane = threadIdx.x & 31;
    const int L    = lane & 15;
    const bool hi  = lane >= 16;

    // Uniform camera matrix / translation (scalar loads).
    const float r00 = rot[0], r01 = rot[1], r02 = rot[2];
    const float r10 = rot[3], r11 = rot[4], r12 = rot[5];
    const float r20 = rot[6], r21 = rot[7], r22 = rot[8];
    const float t0 = tran[0], t1 = tran[1], t2 = tran[2];

    // My point's position.
    const float px = position[3 * tid + 0];
    const float py = position[3 * tid + 1];
    const float pz = position[3 * tid + 2];

    // A (16x4 f32 layout, ISA 7.12.2): lanes 0-15 v0=K0,v1=K1; lanes 16-31 v0=K2,v1=K3.
    // A[m][k] = rot[m][k] (m<3), A[m][3] = tran[m]; zero elsewhere.
    const float a0 = hi ? ((L == 0) ? r02 : (L == 1) ? r12 : (L == 2) ? r22 : 0.0f)
                        : ((L == 0) ? r00 : (L == 1) ? r10 : (L == 2) ? r20 : 0.0f);
    const float a1 = hi ? ((L == 0) ? t0  : (L == 1) ? t1  : (L == 2) ? t2  : 0.0f)
                        : ((L == 0) ? r01 : (L == 1) ? r11 : (L == 2) ? r21 : 0.0f);

    // B (4x16 f32): column n on lane n; lanes 0-15 v0=K0(x),v1=K1(y);
    // lanes 16-31 v0=K2(z), v1=K3(=1, homogeneous -> adds tran).
    const float zOfL   = __shfl(pz, L, 32);       // z of point base+L
    const float xOfHiL = __shfl(px, L + 16, 32);  // x of point base+16+L
    const float yOfHiL = __shfl(py, L + 16, 32);  // y of point base+16+L

    v2f A;  A[0] = a0;  A[1] = a1;
    v2f B1; B1[0] = hi ? zOfL : px;     B1[1] = hi ? 1.0f : py;      // points base+0..15
    v2f B2; B2[0] = hi ? pz   : xOfHiL; B2[1] = hi ? 1.0f : yOfHiL;  // points base+16..31
    v8f C = {0.0f, 0.0f, 0.0f, 0.0f, 0.0f, 0.0f, 0.0f, 0.0f};

    // D[m][n] = pos_cam[n][m]; lane n (0-15) gets x,y,z in D[0..2].
    v8f D1 = __builtin_amdgcn_wmma_f32_16x16x4_f32(false, A, false, B1,
                                                   (short)0, C, false, false);
    v8f D2 = __builtin_amdgcn_wmma_f32_16x16x4_f32(false, A, false, B2,
                                                   (short)0, C, false, false);

    // Fan D2's results (held in lanes 0-15) out to lanes 16-31.
    const float x2 = __shfl(D2[0], L, 32);
    const float y2 = __shfl(D2[1], L, 32);
    const float z2 = __shfl(D2[2], L, 32);

    const float xc = hi ? x2 : D1[0];
    const float yc = hi ? y2 : D1[1];
    const float zc = hi ? z2 : D1[2];

    gs_finish(tid, xc, yc, zc,
              r00, r01, r02, r10, r11, r12, r20, r21, r22,
              rgb, opacity, quat, scale, out_pos, out_cov, out_rgb, out_op);
}

// Scalar tail for any remainder not a multiple of 128 (not hit for N=2M).
__global__ __launch_bounds__(128) void gs_fwd_tail(
    const float* __restrict__ position,
    const float* __restrict__ rgb,
    const float* __restrict__ opacity,
    const float* __restrict__ quat,
    const float* __restrict__ scale,
    const float* __restrict__ rot,
    const float* __restrict__ tran,
    float* __restrict__ out_pos,
    float* __restrict__ out_cov,
    float* __restrict__ out_rgb,
    float* __restrict__ out_op,
    int start, int N)
{
    const int tid = start + blockIdx.x * 128 + threadIdx.x;
    if (tid >= N) return;

    const float r00 = rot[0], r01 = rot[1], r02 = rot[2];
    const float r10 = rot[3], r11 = rot[4], r12 = rot[5];
    const float r20 = rot[6], r21 = rot[7], r22 = rot[8];
    const float t0 = tran[0], t1 = tran[1], t2 = tran[2];

    const float px = position[3 * tid + 0];
    const float py = position[3 * tid + 1];
    const float pz = position[3 * tid + 2];

    const float xc = px * r00 + py * r01 + pz * r02 + t0;
    const float yc = px * r10 + py * r11 + pz * r12 + t1;
    const float zc = px * r20 + py * r21 + pz * r22 + t2;

    gs_finish(tid, xc, yc, zc,
              r00, r01, r02, r10, r11, r12, r20, r21, r22,
              rgb, opacity, quat, scale, out_pos, out_cov, out_rgb, out_op);
}

extern "C" void kernel_launch(void* const* d_in, const int* in_sizes, int n_in,
                              void* d_out, int out_size, void* d_ws, size_t ws_size,
                              hipStream_t stream) {
    (void)n_in; (void)out_size; (void)d_ws; (void)ws_size;

    const float* position = (const float*)d_in[0];  // (N,3)
    const float* rgb      = (const float*)d_in[1];  // (N,3)
    const float* opacity  = (const float*)d_in[2];  // (N,1)
    const float* quat     = (const float*)d_in[3];  // (N,4)
    const float* scale    = (const float*)d_in[4];  // (N,3)
    const float* rot      = (const float*)d_in[5];  // (3,3)
    const float* tran     = (const float*)d_in[6];  // (3,)

    const int N = in_sizes[0] / 3;
    if (N <= 0) return;

    float* out      = (float*)d_out;
    float* out_pos  = out;                    // N*3
    float* out_cov  = out + (size_t)3 * N;    // N*4
    float* out_rgb  = out + (size_t)7 * N;    // N*3
    float* out_op   = out + (size_t)10 * N;   // N*1

    const int main_pts = (N / 128) * 128;
    if (main_pts > 0) {
        gs_fwd_wmma<<<main_pts / 128, 128, 0, stream>>>(
            position, rgb, opacity, quat, scale, rot, tran,
            out_pos, out_cov, out_rgb, out_op);
    }
    const int rem = N - main_pts;
    if (rem > 0) {
        gs_fwd_tail<<<(rem + 127) / 128, 128, 0, stream>>>(
            position, rgb, opacity, quat, scale, rot, tran,
            out_pos, out_cov, out_rgb, out_op, main_pts, N);
    }
}